// MoEModel_47244640256353
// MI455X (gfx1250) — compile-verified
//
#include <hip/hip_runtime.h>
#include <math.h>

typedef __attribute__((ext_vector_type(2))) float v2f;
typedef __attribute__((ext_vector_type(8))) float v8f;

#define NB   8192
#define FEAT 1600
#define NEXP 5
#define NCLS 10

#define H1_N   (13 * 13 * 32)   // 5408 valid floats
#define H1_PAD 928              // zeroed pad region; max koff = 926 (+1 for .y)

static __device__ __forceinline__ v8f wmma_f32_k4(v2f a, v2f b, v8f c) {
  // D = A(16x4 f32) * B(4x16 f32) + C(16x16 f32), exact f32 on matrix pipe
  return __builtin_amdgcn_wmma_f32_16x16x4_f32(false, a, false, b, (short)0, c,
                                               false, false);
}

// K-pair interleaved index: element (k, n) of a [K x 64] B matrix lives at
// (k>>1)*128 + n*2 + (k&1)  -> the (k, k+1) pair for a given n is one b64.
static __device__ __forceinline__ int bidx(int k, int n) {
  return ((k >> 1) << 7) + (n << 1) + (k & 1);
}

// ---------------------------------------------------------------------------
// Pack gate_w [1600,5] and exp_w [5,1600,10] into Wc [1600,64] (interleaved):
//   col n<5 : gate_w[k][n];  col 5..54 : exp_w[(n-5)/10][k][(n-5)%10]; else 0
// ---------------------------------------------------------------------------
__global__ void pack_wc_kernel(const float* __restrict__ gate_w,
                               const float* __restrict__ exp_w,
                               float* __restrict__ wc) {
  int idx = blockIdx.x * 256 + threadIdx.x;
  if (idx >= FEAT * 64) return;
  int k = idx >> 6;
  int n = idx & 63;
  float v = 0.f;
  if (n < 5) {
    v = gate_w[k * NEXP + n];
  } else if (n < 55) {
    int e = (n - 5) / 10, h = (n - 5) % 10;
    v = exp_w[((size_t)e * FEAT + k) * NCLS + h];
  }
  wc[bidx(k, n)] = v;
}

// Pre-interleave conv2 weights [288,64] once (HWIO flat rows are K-major).
__global__ void pack_w2_kernel(const float* __restrict__ w2,
                               float* __restrict__ w2i) {
  int idx = blockIdx.x * 256 + threadIdx.x;
  if (idx >= 288 * 64) return;
  int k = idx >> 6;
  int n = idx & 63;
  w2i[bidx(k, n)] = w2[idx];
}

// ---------------------------------------------------------------------------
// One workgroup (128 threads = 4 waves) per image:
//   conv1(3x3,1->32)+ReLU+pool2 -> h1s[13][13][32]   (VALU, LDS)
//   conv2(3x3,32->64) as implicit GEMM M=128(121)xN=64xK=288 via
//   v_wmma_f32_16x16x4_f32, +bias+ReLU -> c2s, pool2 -> feat[b][1600]
// Pad rows (m>=121) read from a zeroed LDS region: no per-iteration cndmask.
// ---------------------------------------------------------------------------
__global__ __launch_bounds__(128) void conv_feat_kernel(
    const float* __restrict__ x, const float* __restrict__ w1,
    const float* __restrict__ b1, const float* __restrict__ w2i,
    const float* __restrict__ b2, float* __restrict__ feat) {
  __shared__ __align__(16) float xs[28 * 28];          //  3136 B
  __shared__ __align__(16) float w1s[3 * 3 * 32];      //  1152 B
  __shared__ float b1s[32];
  __shared__ float b2s[64];
  __shared__ __align__(16) float h1s[H1_N + H1_PAD];   // 25344 B (incl zero pad)
  __shared__ __align__(16) float w2s[288 * 64];        // 73728 B (interleaved)
  __shared__ __align__(16) float c2s[121 * 64];        // 30976 B

  const int b   = blockIdx.x;
  const int tid = threadIdx.x;

  // ---- stage image + weights into LDS; zero the A-pad region ----
  {
    const float4* xsrc = (const float4*)(x + (size_t)b * 784);
    float4* xdst = (float4*)xs;
    for (int i = tid; i < 196; i += 128) xdst[i] = xsrc[i];
    const float4* wsrc = (const float4*)w2i;  // already interleaved
    float4* wdst = (float4*)w2s;
    for (int i = tid; i < 18432 / 4; i += 128) wdst[i] = wsrc[i];
    for (int i = tid; i < 288; i += 128) w1s[i] = w1[i];
    for (int i = tid; i < H1_PAD; i += 128) h1s[H1_N + i] = 0.f;
    if (tid < 32) b1s[tid] = b1[tid];
    if (tid < 64) b2s[tid] = b2[tid];
  }
  __syncthreads();

  // ---- conv1 + ReLU + 2x2 maxpool -> h1s[13][13][32] ----
  for (int idx = tid; idx < H1_N; idx += 128) {
    int c = idx & 31;
    int p = idx >> 5;
    int py = p / 13, px = p % 13;
    float m = 0.f;  // max(0, conv...) == pool(relu(conv)) since relu monotone
    #pragma unroll
    for (int dy = 0; dy < 2; ++dy) {
      #pragma unroll
      for (int dx = 0; dx < 2; ++dx) {
        int oy = 2 * py + dy, ox = 2 * px + dx;
        float acc = b1s[c];
        #pragma unroll
        for (int ky = 0; ky < 3; ++ky)
          #pragma unroll
          for (int kx = 0; kx < 3; ++kx)
            acc += xs[(oy + ky) * 28 + (ox + kx)] * w1s[(ky * 3 + kx) * 32 + c];
        m = fmaxf(m, acc);
      }
    }
    h1s[idx] = m;
  }
  __syncthreads();

  // ---- conv2 implicit GEMM via WMMA f32 16x16x4 ----
  const int lane = tid & 31;
  const int wave = tid >> 5;
  const int lm = lane & 15;  // M (for A) / N (for B) within 16-subtile
  const int lh = lane >> 4;  // lane half selects K pair

  v8f acc[2][4];
  {
    v8f z = {0.f, 0.f, 0.f, 0.f, 0.f, 0.f, 0.f, 0.f};
    #pragma unroll
    for (int s = 0; s < 2; ++s)
      #pragma unroll
      for (int t = 0; t < 4; ++t) acc[s][t] = z;
  }

  // per-lane row bases (loop-invariant); pad rows point at the zero region
  int rowbase[2];
  #pragma unroll
  for (int s = 0; s < 2; ++s) {
    int m = wave * 32 + s * 16 + lm;
    int oy = m / 11, ox = m % 11;
    rowbase[s] = (m < 121) ? (oy * 13 + ox) * 32 : H1_N;
  }

  for (int k0 = 0; k0 < 288; k0 += 4) {
    int k = k0 + 2 * lh;                 // this lane-half's first K
    int ky = k / 96;
    int kr = k - ky * 96;
    int kx = kr >> 5;
    int ci = kr & 31;                    // even; ci,ci+1 stay in same 32-block
    int koff = (ky * 13 + kx) * 32 + ci; // max 926; pad region covers it

    // B operands: 4 N-subtiles, one aligned b64 each (interleaved layout)
    int kh = (k0 >> 1) + lh;             // == k>>1 (k even)
    v2f bb[4];
    #pragma unroll
    for (int t = 0; t < 4; ++t)
      bb[t] = *(const v2f*)&w2s[(kh << 7) + ((t * 16 + lm) << 1)];

    // A operands: 2 M-subtiles, one b64 load each (pad rows read zeros)
    v2f aa[2];
    #pragma unroll
    for (int s = 0; s < 2; ++s) aa[s] = *(const v2f*)&h1s[rowbase[s] + koff];

    #pragma unroll
    for (int s = 0; s < 2; ++s)
      #pragma unroll
      for (int t = 0; t < 4; ++t) acc[s][t] = wmma_f32_k4(aa[s], bb[t], acc[s][t]);
  }

  // ---- epilogue: bias + ReLU, D-layout (VGPR r, lanehalf h -> M = r + 8h) ----
  #pragma unroll
  for (int s = 0; s < 2; ++s) {
    int mbase = wave * 32 + s * 16 + 8 * lh;
    #pragma unroll
    for (int t = 0; t < 4; ++t) {
      int n = t * 16 + lm;
      #pragma unroll
      for (int r = 0; r < 8; ++r) {
        int m = mbase + r;
        if (m < 121) c2s[m * 64 + n] = fmaxf(acc[s][t][r] + b2s[n], 0.f);
      }
    }
  }
  __syncthreads();

  // ---- 2x2 maxpool 11x11 -> 5x5, flatten to feat[b][1600] ----
  for (int idx = tid; idx < 1600; idx += 128) {
    int c = idx & 63;
    int p = idx >> 6;
    int py = p / 5, px = p % 5;
    int o = (2 * py * 11 + 2 * px) * 64 + c;
    float m0 = fmaxf(c2s[o], c2s[o + 64]);
    float m1 = fmaxf(c2s[o + 11 * 64], c2s[o + 11 * 64 + 64]);
    feat[(size_t)b * FEAT + idx] = fmaxf(m0, m1);
  }
}

// ---------------------------------------------------------------------------
// logits[8192,64] = feat[8192,1600] @ Wc[1600,64]   (WMMA f32 16x16x4)
// Workgroup = 128 threads, M-tile 128; K panels of 160 staged in LDS.
// ---------------------------------------------------------------------------
__global__ __launch_bounds__(128) void moe_logits_kernel(
    const float* __restrict__ feat, const float* __restrict__ wc,
    float* __restrict__ logits) {
  __shared__ __align__(16) float bpan[160 * 64];  // 40 KB (interleaved)

  const int tid = threadIdx.x;
  const int lane = tid & 31, wave = tid >> 5;
  const int lm = lane & 15, lh = lane >> 4;
  const int Mbase = blockIdx.x * 128;

  v8f acc[2][4];
  {
    v8f z = {0.f, 0.f, 0.f, 0.f, 0.f, 0.f, 0.f, 0.f};
    #pragma unroll
    for (int s = 0; s < 2; ++s)
      #pragma unroll
      for (int t = 0; t < 4; ++t) acc[s][t] = z;
  }

  for (int kp = 0; kp < FEAT; kp += 160) {
    __syncthreads();
    {
      const float4* src = (const float4*)(wc + (size_t)kp * 64);
      float4* dst = (float4*)bpan;
      for (int i = tid; i < (160 * 64) / 4; i += 128) dst[i] = src[i];
    }
    __syncthreads();

    for (int kk = 0; kk < 160; kk += 4) {
      int kb = kk + 2 * lh;
      int kh = (kk >> 1) + lh;  // == kb>>1
      v2f bb[4];
      #pragma unroll
      for (int t = 0; t < 4; ++t)
        bb[t] = *(const v2f*)&bpan[(kh << 7) + ((t * 16 + lm) << 1)];
      v2f aa[2];
      #pragma unroll
      for (int s = 0; s < 2; ++s) {
        int m = Mbase + wave * 32 + s * 16 + lm;
        aa[s] = *(const v2f*)&feat[(size_t)m * FEAT + kp + kb];
      }
      #pragma unroll
      for (int s = 0; s < 2; ++s)
        #pragma unroll
        for (int t = 0; t < 4; ++t) acc[s][t] = wmma_f32_k4(aa[s], bb[t], acc[s][t]);
    }
  }

  #pragma unroll
  for (int s = 0; s < 2; ++s) {
    int mbase = Mbase + wave * 32 + s * 16 + 8 * lh;
    #pragma unroll
    for (int t = 0; t < 4; ++t) {
      int n = t * 16 + lm;
      #pragma unroll
      for (int r = 0; r < 8; ++r) logits[(size_t)(mbase + r) * 64 + n] = acc[s][t][r];
    }
  }
}

// ---------------------------------------------------------------------------
// Per-row: gate softmax(5), stable top-3, weighted expert combine, softmax(10)
// ---------------------------------------------------------------------------
__global__ void finalize_kernel(const float* __restrict__ logits,
                                const float* __restrict__ gate_b,
                                const float* __restrict__ exp_b,
                                float* __restrict__ out) {
  int b = blockIdx.x * 256 + threadIdx.x;
  if (b >= NB) return;
  const float* L = logits + (size_t)b * 64;

  float g[NEXP];
  float gm = -1e30f;
  #pragma unroll
  for (int e = 0; e < NEXP; ++e) {
    g[e] = L[e] + gate_b[e];
    gm = fmaxf(gm, g[e]);
  }
  float gs = 0.f;
  #pragma unroll
  for (int e = 0; e < NEXP; ++e) {
    g[e] = expf(g[e] - gm);
    gs += g[e];
  }
  float inv = 1.f / gs;
  #pragma unroll
  for (int e = 0; e < NEXP; ++e) g[e] *= inv;

  float comb[NCLS];
  #pragma unroll
  for (int h = 0; h < NCLS; ++h) comb[h] = 0.f;

  bool used[NEXP] = {false, false, false, false, false};
  for (int kk = 0; kk < 3; ++kk) {  // strict '>' => earliest index on ties (lax.top_k)
    int best = 0;
    float bv = -1.f;
    #pragma unroll
    for (int e = 0; e < NEXP; ++e) {
      if (!used[e] && g[e] > bv) { bv = g[e]; best = e; }
    }
    used[best] = true;
    #pragma unroll
    for (int h = 0; h < NCLS; ++h)
      comb[h] += bv * (L[5 + best * NCLS + h] + exp_b[best * NCLS + h]);
  }

  float cm = comb[0];
  #pragma unroll
  for (int h = 1; h < NCLS; ++h) cm = fmaxf(cm, comb[h]);
  float cs = 0.f;
  float ex[NCLS];
  #pragma unroll
  for (int h = 0; h < NCLS; ++h) {
    ex[h] = expf(comb[h] - cm);
    cs += ex[h];
  }
  float cinv = 1.f / cs;
  #pragma unroll
  for (int h = 0; h < NCLS; ++h) out[(size_t)b * NCLS + h] = ex[h] * cinv;
}

// ---------------------------------------------------------------------------
extern "C" void kernel_launch(void* const* d_in, const int* in_sizes, int n_in,
                              void* d_out, int out_size, void* d_ws, size_t ws_size,
                              hipStream_t stream) {
  const float* x  = (const float*)d_in[0];
  const float* w1 = (const float*)d_in[1];
  const float* b1 = (const float*)d_in[2];
  const float* w2 = (const float*)d_in[3];
  const float* b2 = (const float*)d_in[4];
  const float* gw = (const float*)d_in[5];
  const float* gb = (const float*)d_in[6];
  const float* ew = (const float*)d_in[7];
  const float* eb = (const float*)d_in[8];
  float* out = (float*)d_out;

  // workspace layout (floats): feat | wc | logits | w2i  (~55 MB total)
  float* feat   = (float*)d_ws;                    // 8192*1600
  float* wc     = feat + (size_t)NB * FEAT;        // 1600*64 (interleaved)
  float* logits = wc + (size_t)FEAT * 64;          // 8192*64
  float* w2i    = logits + (size_t)NB * 64;        // 288*64  (interleaved)

  pack_wc_kernel<<<(FEAT * 64 + 255) / 256, 256, 0, stream>>>(gw, ew, wc);
  pack_w2_kernel<<<(288 * 64 + 255) / 256, 256, 0, stream>>>(w2, w2i);
  conv_feat_kernel<<<NB, 128, 0, stream>>>(x, w1, b1, w2i, b2, feat);
  moe_logits_kernel<<<NB / 128, 128, 0, stream>>>(feat, wc, logits);
  finalize_kernel<<<(NB + 255) / 256, 256, 0, stream>>>(logits, gb, eb, out);
}